// xERTE_3332894622230
// MI455X (gfx1250) — compile-verified
//
#include <hip/hip_runtime.h>
#include <cfloat>

#define E_TOTAL (64 * 4096)
#define DIMS    256
#define BATCH   64
#define EPER    4096
#define TOPK    50
#define ROWP    132      // 128 dims + 4 pad dwords -> bank (4m + k) % 64, conflict-free
#define NWAVES  4        // waves per block in score kernel

typedef __attribute__((ext_vector_type(2))) float v2f;
typedef __attribute__((ext_vector_type(4))) float v4f;
typedef __attribute__((ext_vector_type(8))) float v8f;

// ---------------------------------------------------------------------------
// Kernel 1: target_score for all E edges.
// One wave == one 16-edge tile. Coalesced NT global streaming -> LDS staging
// -> V_WMMA_F32_16X16X4_F32 reduction (B = ones, C chained).
// ---------------------------------------------------------------------------
__global__ __launch_bounds__(32 * NWAVES) void score_kernel(
    const float* __restrict__ step_w,       // 3   (weights)
    const float* __restrict__ sample_loss,  // E
    const float* __restrict__ vj,           // E*256
    const float* __restrict__ rel,          // E*256
    const float* __restrict__ q_src,        // 64*256
    const float* __restrict__ q_rel,        // 64*256
    float* __restrict__ out_score)          // E
{
    __shared__ alignas(16) float smem[NWAVES][2][16 * ROWP];

    const int wave = threadIdx.x >> 5;
    const int lane = threadIdx.x & 31;
    const int tile = blockIdx.x * NWAVES + wave;   // 16 edges per tile
    const int base = tile * 16;
    const int qidx = base >> 12;                   // base / 4096 (constant per tile)

    float* tbuf = smem[wave][0];                   // |q_src + rel - vj|
    float* rbuf = smem[wave][1];                   // |q_rel - rel|

    // WMMA 16x4 f32 A-layout coordinates for this lane
    const int m    = lane & 15;                    // edge-in-tile (A row)
    const int koff = (lane >> 4) << 1;             // lanes 0-15 -> K 0,1 ; 16-31 -> K 2,3

    v8f acc_t = {};
    v8f acc_r = {};
    const v2f bones = {1.0f, 1.0f};                // B = ones: any D column = row-sum

    #pragma unroll
    for (int h = 0; h < 2; ++h) {
        // ---- Phase A: coalesced stream + abs-diff -> LDS ----
        const v4f qs = ((const v4f*)(q_src + qidx * DIMS + h * 128))[lane];
        const v4f qr = ((const v4f*)(q_rel + qidx * DIMS + h * 128))[lane];

        #pragma unroll 4
        for (int mm = 0; mm < 16; ++mm) {
            const size_t e = (size_t)(base + mm);
            v4f r4 = __builtin_nontemporal_load((const v4f*)(rel + e * DIMS + h * 128) + lane);
            v4f v4 = __builtin_nontemporal_load((const v4f*)(vj  + e * DIMS + h * 128) + lane);
            v4f t4, l4;
            t4.x = fabsf(qs.x + r4.x - v4.x);  l4.x = fabsf(qr.x - r4.x);
            t4.y = fabsf(qs.y + r4.y - v4.y);  l4.y = fabsf(qr.y - r4.y);
            t4.z = fabsf(qs.z + r4.z - v4.z);  l4.z = fabsf(qr.z - r4.z);
            t4.w = fabsf(qs.w + r4.w - v4.w);  l4.w = fabsf(qr.w - r4.w);
            ((v4f*)(tbuf + mm * ROWP))[lane] = t4;   // dword mm*132 + 4*lane
            ((v4f*)(rbuf + mm * ROWP))[lane] = l4;
        }
        __syncthreads();

        // ---- Phase B: WMMA reduction over this half's 128 dims ----
        #pragma unroll 8
        for (int kk = 0; kk < 128; kk += 4) {
            v2f at = *(const v2f*)(tbuf + m * ROWP + kk + koff);   // ds_load_b64
            v2f ar = *(const v2f*)(rbuf + m * ROWP + kk + koff);
            acc_t = __builtin_amdgcn_wmma_f32_16x16x4_f32(
                false, at, false, bones, (short)0, acc_t, false, false);
            acc_r = __builtin_amdgcn_wmma_f32_16x16x4_f32(
                false, ar, false, bones, (short)0, acc_r, false, false);
        }
        __syncthreads();
    }

    const float w0 = fabsf(step_w[0]);
    const float w1 = fabsf(step_w[1]);
    const float w2 = fabsf(step_w[2]);

    // D layout: lane 0 VGPR r -> M=r ; lane 16 VGPR r -> M=8+r (any column N)
    if (lane == 0 || lane == 16) {
        const int mb = (lane == 0) ? 0 : 8;
        #pragma unroll
        for (int r = 0; r < 8; ++r) {
            const int e = base + mb + r;
            out_score[e] = -(w0 * acc_r[r] + w1 * sample_loss[e] + w2 * acc_t[r]);
        }
    }
}

// ---------------------------------------------------------------------------
// Kernel 2: per-batch top-50 (iterative argmax, low-index tie-break like
// lax.top_k), then emit orig_indices + gathered pruned_edges as floats.
// ---------------------------------------------------------------------------
__global__ __launch_bounds__(256) void topk_kernel(
    const float* __restrict__ score,     // E (written by score_kernel)
    const int*   __restrict__ edges,     // E*9
    float* __restrict__ out_pruned,      // 3200*9
    float* __restrict__ out_idx)         // 3200
{
    __shared__ float sc[EPER];
    __shared__ float rv[256];
    __shared__ int   ri[256];
    __shared__ int   sel[TOPK];

    const int b = blockIdx.x;
    const int t = threadIdx.x;

    for (int i = t; i < EPER; i += 256) sc[i] = score[b * EPER + i];
    __syncthreads();

    for (int r = 0; r < TOPK; ++r) {
        float bv = -FLT_MAX; int bi = 0x7fffffff;
        for (int i = t; i < EPER; i += 256) {
            float v = sc[i];
            if (v > bv || (v == bv && i < bi)) { bv = v; bi = i; }
        }
        rv[t] = bv; ri[t] = bi;
        __syncthreads();
        for (int s = 128; s > 0; s >>= 1) {
            if (t < s) {
                float ov = rv[t + s]; int oi = ri[t + s];
                if (ov > rv[t] || (ov == rv[t] && oi < ri[t])) { rv[t] = ov; ri[t] = oi; }
            }
            __syncthreads();
        }
        if (t == 0) { sel[r] = ri[0]; sc[ri[0]] = -FLT_MAX; }
        __syncthreads();
    }

    if (t < TOPK)
        out_idx[b * TOPK + t] = (float)(b * EPER + sel[t]);
    if (t < TOPK * 9) {
        const int r = t / 9, c = t % 9;
        const int orig = b * EPER + sel[r];
        out_pruned[(b * TOPK + r) * 9 + c] = (float)edges[(size_t)orig * 9 + c];
    }
}

// ---------------------------------------------------------------------------
// Kernel 3: visited_nodes_reg passthrough (1 MB, NT vector copy).
// ---------------------------------------------------------------------------
__global__ __launch_bounds__(256) void copy_kernel(
    const v4f* __restrict__ src, v4f* __restrict__ dst, int n4)
{
    const int i = blockIdx.x * blockDim.x + threadIdx.x;
    if (i < n4) {
        v4f v = __builtin_nontemporal_load(src + i);
        __builtin_nontemporal_store(v, dst + i);
    }
}

extern "C" void kernel_launch(void* const* d_in, const int* in_sizes, int n_in,
                              void* d_out, int out_size, void* d_ws, size_t ws_size,
                              hipStream_t stream) {
    (void)in_sizes; (void)n_in; (void)d_ws; (void)ws_size; (void)out_size;

    const float* step_w      = (const float*)d_in[0];   // (3,1)
    const float* visited     = (const float*)d_in[1];   // (E,)
    const float* sample_loss = (const float*)d_in[3];   // (E,)
    const float* vj          = (const float*)d_in[5];   // (E,256)
    const float* rel         = (const float*)d_in[6];   // (E,256)
    const float* qsrc        = (const float*)d_in[7];   // (64,256)
    const float* qrel        = (const float*)d_in[8];   // (64,256)
    const int*   edges       = (const int*)d_in[9];     // (E,9)

    float* out       = (float*)d_out;
    float* out_score = out;                                    // E
    float* out_vis   = out + E_TOTAL;                          // E
    float* out_pe    = out + 2 * E_TOTAL;                      // 3200*9
    float* out_idx   = out + 2 * E_TOTAL + BATCH * TOPK * 9;   // 3200

    score_kernel<<<E_TOTAL / (16 * NWAVES), 32 * NWAVES, 0, stream>>>(
        step_w, sample_loss, vj, rel, qsrc, qrel, out_score);

    copy_kernel<<<(E_TOTAL / 4 + 255) / 256, 256, 0, stream>>>(
        (const v4f*)visited, (v4f*)out_vis, E_TOTAL / 4);

    topk_kernel<<<BATCH, 256, 0, stream>>>(out_score, edges, out_pe, out_idx);
}